// MaskRCNN_87900800680162
// MI455X (gfx1250) — compile-verified
//
#include <hip/hip_runtime.h>

// ROI-Align (FPN, Mask R-CNN) for MI455X / gfx1250.
//
// Strategy: x-interpolation expressed as dense 16x16x16 f32 GEMM
// (4 chained V_WMMA_F32_16X16X4_F32) with the 7x14 interpolation weight
// matrix as A (shared across channels), gathered feature rows as B
// (16 channels in lanes). y-interpolation is 2 VALU FMAs. Output staged
// per-wave in LDS and written with fully coalesced contiguous stores.

typedef float v2f __attribute__((ext_vector_type(2)));
typedef float v8f __attribute__((ext_vector_type(8)));

#define POOL 7
#define NCH  256

__global__ __launch_bounds__(256) void roi_align_wmma_kernel(
    const float* __restrict__ boxes,
    const float* __restrict__ p2, const float* __restrict__ p3,
    const float* __restrict__ p4, const float* __restrict__ p5,
    float* __restrict__ out)
{
    __shared__ float s_wy[POOL], s_my[POOL];
    __shared__ float s_wx[POOL], s_mx[POOL];
    __shared__ int   s_yrow[16];   // 14 clamped y indices (y0,y1 per sample row) + pad
    __shared__ int   s_xc[16];     // 14 clamped x indices (x0,x1 per sample col) + pad
    __shared__ float s_buf[8][16 * 49];  // per-wave staging: [c_local][py*7+px]

    const int n    = blockIdx.x;
    const int tid  = threadIdx.x;
    const int lane = tid & 31;
    const int wave = tid >> 5;

    // ---- box + FPN level (redundant per thread, all scalar-uniform) ----
    const float y1 = boxes[n * 4 + 0];
    const float x1 = boxes[n * 4 + 1];
    const float y2 = boxes[n * 4 + 2];
    const float x2 = boxes[n * 4 + 3];
    const float bh = y2 - y1, bw = x2 - x1;
    // roi_level = clip(round(4 + log2(sqrt(h*w) / (224/1024))), 2, 5)
    float lvl_f = 4.0f + log2f(sqrtf(bh * bw) / 0.21875f);
    int level = (int)rintf(lvl_f);            // round-half-to-even, matches jnp.round
    level = min(5, max(2, level));

    const float* fm;
    int H;
    switch (level) {
        case 2:  fm = p2; H = 256; break;
        case 3:  fm = p3; H = 128; break;
        case 4:  fm = p4; H = 64;  break;
        default: fm = p5; H = 32;  break;
    }
    const int W  = H;
    const int HW = H * W;

    // ---- sampling geometry (TF crop_and_resize semantics) ----
    if (tid < POOL) {
        const float t = (float)tid / 6.0f;

        const float ys  = (y1 + bh * t) * (float)(H - 1);
        const float y0f = floorf(ys);
        int yi0 = (int)y0f;
        yi0 = min(H - 1, max(0, yi0));
        const int yi1 = min(H - 1, yi0 + 1);
        s_wy[tid]          = ys - y0f;
        s_my[tid]          = (ys >= 0.0f && ys <= (float)(H - 1)) ? 1.0f : 0.0f;
        s_yrow[2 * tid]     = yi0;
        s_yrow[2 * tid + 1] = yi1;

        const float xs  = (x1 + bw * t) * (float)(W - 1);
        const float x0f = floorf(xs);
        int xi0 = (int)x0f;
        xi0 = min(W - 1, max(0, xi0));
        const int xi1 = min(W - 1, xi0 + 1);
        s_wx[tid]         = xs - x0f;
        s_mx[tid]         = (xs >= 0.0f && xs <= (float)(W - 1)) ? 1.0f : 0.0f;
        s_xc[2 * tid]     = xi0;
        s_xc[2 * tid + 1] = xi1;
        if (tid == 6) {  // pad K-columns 14,15 with a valid (finite) index
            s_xc[14] = xi1;  s_xc[15] = xi1;
            s_yrow[14] = yi1; s_yrow[15] = yi1;
        }
    }
    __syncthreads();

    // ---- build A (x-interp weights), 16x16 padded, held in 8 VGPRs ----
    // A[px, 2px]   = (1-wx[px])*mx[px]
    // A[px, 2px+1] =    wx[px] *mx[px],  zero elsewhere / for px >= 7.
    // 32-bit 16x4 A layout: lanes 0-15 = M rows; VGPR0 holds K={0,2}, VGPR1 K={1,3}.
    const int pxl   = lane & 15;           // M-row (= px) / channel-in-tile
    const int khalf = (lane >> 4) << 1;    // 0 for lanes 0-15, 2 for lanes 16-31
    const float wxv = (pxl < POOL) ? s_wx[pxl] : 0.0f;
    const float mxv = (pxl < POOL) ? s_mx[pxl] : 0.0f;
    const float a_lo = (1.0f - wxv) * mxv;
    const float a_hi = wxv * mxv;

    v2f A[4];
#pragma unroll
    for (int s = 0; s < 4; ++s) {
        const int j0 = 4 * s + khalf;
        const int j1 = j0 + 1;
        A[s][0] = (j0 == 2 * pxl) ? a_lo : (j0 == 2 * pxl + 1) ? a_hi : 0.0f;
        A[s][1] = (j1 == 2 * pxl) ? a_lo : (j1 == 2 * pxl + 1) ? a_hi : 0.0f;
    }

    // ---- main loop: each wave handles two 16-channel tiles ----
    for (int t = wave; t < 16; t += 8) {
        const int c0 = t * 16;
        const float* fmc = fm + (size_t)(c0 + pxl) * HW;  // this lane's channel plane
        float* buf = s_buf[wave];

        for (int py = 0; py < POOL; ++py) {
            v8f q[2];
#pragma unroll
            for (int rr = 0; rr < 2; ++rr) {
                const float* rowp = fmc + s_yrow[2 * py + rr] * W;
                v8f acc = {};
#pragma unroll
                for (int s = 0; s < 4; ++s) {
                    const int j0 = 4 * s + khalf;
                    // B 4x16 f32 layout: VGPR0 = K rows {0,2}, VGPR1 = K rows {1,3},
                    // N (channels) across lanes 0-15 / 16-31.
                    v2f b;
                    b[0] = rowp[s_xc[j0]];
                    b[1] = rowp[s_xc[j0 + 1]];
                    acc = __builtin_amdgcn_wmma_f32_16x16x4_f32(
                        /*neg_a=*/false, A[s], /*neg_b=*/false, b,
                        /*c_mod=*/(short)0, acc,
                        /*reuse_a=*/false, /*reuse_b=*/false);
                }
                q[rr] = acc;
            }
            // y-blend + OOB mask, stage into LDS.
            // C/D layout: VGPR v holds M=v (lanes 0-15) -> px = v for lanes < 16.
            const float wyv = s_wy[py], myv = s_my[py];
            const float w0 = (1.0f - wyv) * myv;
            const float w1 = wyv * myv;
            if (lane < 16) {
#pragma unroll
                for (int v = 0; v < POOL; ++v) {
                    buf[pxl * 49 + py * 7 + v] = w0 * q[0][v] + w1 * q[1][v];
                }
            }
        }

        // Coalesced writeback: out[n, c0..c0+16, 0..49) is 784 contiguous floats.
        const size_t obase = (size_t)n * (NCH * 49) + (size_t)c0 * 49;
        for (int i = lane; i < 16 * 49; i += 32) {
            out[obase + i] = buf[i];
        }
    }
}

extern "C" void kernel_launch(void* const* d_in, const int* in_sizes, int n_in,
                              void* d_out, int out_size, void* d_ws, size_t ws_size,
                              hipStream_t stream) {
    const float* boxes = (const float*)d_in[0];
    const float* p2    = (const float*)d_in[1];
    const float* p3    = (const float*)d_in[2];
    const float* p4    = (const float*)d_in[3];
    const float* p5    = (const float*)d_in[4];
    float* out = (float*)d_out;

    (void)in_sizes; (void)n_in; (void)out_size; (void)d_ws; (void)ws_size;

    roi_align_wmma_kernel<<<1000, 256, 0, stream>>>(boxes, p2, p3, p4, p5, out);
}